// PCSNNTorchTopology_9990093930916
// MI455X (gfx1250) — compile-verified
//
#include <hip/hip_runtime.h>

typedef __attribute__((ext_vector_type(2))) float v2f;
typedef __attribute__((ext_vector_type(8))) float v8f;

#define NRG 128
#define NB  8
#define NN  1024
constexpr float TAU     = 0.8f;
constexpr float ALPHA   = 0.05f;
constexpr float HEBB_LR = 1e-5f;

// ---- Phase 1: incoming message sum + x + rbar trace; also zero-pad x rows 8..15
__global__ void k_pre(const float* __restrict__ A,
                      const float* __restrict__ x_inp,
                      const float* __restrict__ rbar_pre,
                      const float* __restrict__ vs_ws,
                      float* __restrict__ x_pad,   // [16][1024], rows 8..15 zero
                      float* __restrict__ rb,      // [8][1024]
                      int i) {
  int idx = blockIdx.x * blockDim.x + threadIdx.x;
  if (idx < NB * NN) {
    float pre = 0.f;
#pragma unroll 4
    for (int r = 0; r < NRG; ++r)
      pre = fmaf(A[r * NRG + i], vs_ws[r * (NB * NN) + idx], pre);
    float x = pre + x_inp[(size_t)i * (NB * NN) + idx];
    x_pad[idx] = x;
    rb[idx] = (1.f - ALPHA) * rbar_pre[(size_t)i * (NB * NN) + idx] + ALPHA * x;
  } else if (idx < 16 * NN) {
    x_pad[idx] = 0.f;   // pad M=8 -> 16 for WMMA
  }
}

// ---- Phase 2: pred_b = x_pad @ W  (M=16(pad), K=1024, N=1024) via v_wmma_f32_16x16x4_f32
// One wave per 16-wide N tile; dual accumulators for ILP; prefetch next region's W into L2.
__global__ void k_gemm_pred(const float* __restrict__ x_pad,
                            const float* __restrict__ W,
                            const float* __restrict__ Wnext,
                            float* __restrict__ pred) {
  int lane = threadIdx.x & 31;
  int wave = (blockIdx.x * blockDim.x + threadIdx.x) >> 5;  // 0..63
  int n0   = wave * 16;
  int half = lane >> 4;     // 0: K pair {k,k+1}; 1: K pair {k+2,k+3}
  int l15  = lane & 15;
  v8f acc0 = {}; v8f acc1 = {};
  for (int k0 = 0; k0 < NN; k0 += 8) {
    int ka = k0 + half * 2;
    int kb = ka + 4;
    v2f a0, b0, a1, b1;
    a0.x = x_pad[l15 * NN + ka];
    a0.y = x_pad[l15 * NN + ka + 1];
    b0.x = W[(size_t)(ka    ) * NN + n0 + l15];
    b0.y = W[(size_t)(ka + 1) * NN + n0 + l15];
    a1.x = x_pad[l15 * NN + kb];
    a1.y = x_pad[l15 * NN + kb + 1];
    b1.x = W[(size_t)(kb    ) * NN + n0 + l15];
    b1.y = W[(size_t)(kb + 1) * NN + n0 + l15];
    acc0 = __builtin_amdgcn_wmma_f32_16x16x4_f32(false, a0, false, b0, (short)0, acc0, false, false);
    acc1 = __builtin_amdgcn_wmma_f32_16x16x4_f32(false, a1, false, b1, (short)0, acc1, false, false);
  }
  v8f acc = acc0 + acc1;
  // Pull next region's 4MB W into L2 (192MB) to hide the serial region chain.
  int tid = blockIdx.x * blockDim.x + threadIdx.x;   // 2048 threads
#pragma unroll
  for (int p = 0; p < 16; ++p)
    __builtin_prefetch(Wnext + ((size_t)tid * 16 + p) * 32, 0, 1);
  if (lane < 16) {  // rows 0..7 of the D tile are the real M=8 outputs
#pragma unroll
    for (int v = 0; v < 8; ++v)
      pred[v * NN + n0 + lane] = acc[v];
  }
}

// ---- Phase 3: leaky updates, spike, ebar trace, v_s scan update, deterministic F partials
__global__ void k_elem(const float* __restrict__ pred,
                       const float* __restrict__ v_b,
                       const float* __restrict__ v_a,
                       const float* __restrict__ ebar_post,
                       float* __restrict__ vs_ws,
                       float* __restrict__ eb,
                       float* __restrict__ sp_out,
                       float* __restrict__ Fpart,   // [NRG][256] wave partial sums
                       int i) {
  int idx = blockIdx.x * blockDim.x + threadIdx.x;
  size_t g = (size_t)i * (NB * NN) + idx;
  float pb = pred[idx];
  float vb = TAU * v_b[g] + (1.f - TAU) * pb;
  float va = TAU * v_a[g];
  float vs = TAU * vs_ws[g] + (1.f - TAU) * (vb - va);
  vs_ws[g]  = vs;                       // Gauss-Seidel: visible to regions j>i
  sp_out[g] = (vs > 0.f) ? 1.f : 0.f;
  eb[idx] = (1.f - ALPHA) * ebar_post[g] + ALPHA * (vs - vb);
  float d = vs - vb;
  float s = d * d;
#pragma unroll
  for (int off = 16; off > 0; off >>= 1)   // wave32 reduction (fixed order)
    s += __shfl_xor(s, off, 32);
  if ((threadIdx.x & 31) == 0) {
    int w = blockIdx.x * (blockDim.x >> 5) + (threadIdx.x >> 5);  // 0..255
    Fpart[i * 256 + w] = s;              // fixed slot -> deterministic
  }
}

// ---- Phase 4: W_new = W + (lr/B) * rb.T @ eb  (rank-8 update, two K=4 WMMA steps)
__global__ void k_wupd(const float* __restrict__ rb,
                       const float* __restrict__ eb,
                       const float* __restrict__ Wi,
                       float* __restrict__ Wo) {
  int lane = threadIdx.x & 31;
  int tile = (blockIdx.x * blockDim.x + threadIdx.x) >> 5;  // 0..4095
  int m0   = (tile >> 6) * 16;
  int n0   = (tile & 63) * 16;
  int half = lane >> 4;
  int l15  = lane & 15;
  v8f acc = {};
#pragma unroll
  for (int k0 = 0; k0 < 8; k0 += 4) {
    int kb = k0 + half * 2;
    v2f a, b;
    a.x = rb[kb * NN + m0 + l15];        // A[m][k] = rb[k][m0+m] (transposed read)
    a.y = rb[(kb + 1) * NN + m0 + l15];
    b.x = eb[kb * NN + n0 + l15];        // B[k][n] = eb[k][n0+n]
    b.y = eb[(kb + 1) * NN + n0 + l15];
    acc = __builtin_amdgcn_wmma_f32_16x16x4_f32(false, a, false, b, (short)0, acc, false, false);
  }
  const float scale = HEBB_LR / (float)NB;
  int mrow = half * 8;
#pragma unroll
  for (int v = 0; v < 8; ++v) {
    size_t off = (size_t)(m0 + mrow + v) * NN + n0 + l15;
    Wo[off] = Wi[off] + scale * acc[v];  // W re-read hits L2
  }
}

// ---- Finalize F: deterministic fixed-order reduction of 128*256 wave partials
__global__ void k_fin(const float* __restrict__ Fpart, float* __restrict__ Fout) {
  __shared__ float red[256];
  int t = threadIdx.x;                   // 256 threads
  float s = 0.f;
#pragma unroll 4
  for (int j = 0; j < NRG; ++j)          // fixed order per thread
    s += Fpart[j * 256 + t];
  red[t] = s;
  __syncthreads();
#pragma unroll
  for (int off = 128; off > 0; off >>= 1) {  // fixed-associativity tree
    if (t < off) red[t] += red[t + off];
    __syncthreads();
  }
  if (t == 0) *Fout = red[0] / (float)((size_t)NRG * NB * NN);
}

extern "C" void kernel_launch(void* const* d_in, const int* in_sizes, int n_in,
                              void* d_out, int out_size, void* d_ws, size_t ws_size,
                              hipStream_t stream) {
  const float* A        = (const float*)d_in[0];
  const float* W_ff     = (const float*)d_in[1];
  const float* v_b      = (const float*)d_in[2];
  const float* v_a      = (const float*)d_in[3];
  const float* v_s      = (const float*)d_in[4];
  const float* rbar_pre = (const float*)d_in[5];
  const float* ebar     = (const float*)d_in[6];
  const float* x_inp    = (const float*)d_in[7];

  float* sp_out = (float*)d_out;                         // [128,8,1024]
  float* F_out  = sp_out + (size_t)NRG * NB * NN;        // [1]
  float* W_out  = F_out + 1;                             // [128,1024,1024]

  float* ws    = (float*)d_ws;
  float* vs_ws = ws;                                     // 1,048,576 floats
  float* x_pad = vs_ws + (size_t)NRG * NB * NN;          // 16,384
  float* rb    = x_pad + 16 * NN;                        // 8,192
  float* eb    = rb + NB * NN;                           // 8,192
  float* pred  = eb + NB * NN;                           // 8,192
  float* Fpart = pred + NB * NN;                         // 128*256 = 32,768

  // mutable copy of v_s (never mutate inputs)
  hipMemcpyAsync(vs_ws, v_s, sizeof(float) * (size_t)NRG * NB * NN,
                 hipMemcpyDeviceToDevice, stream);

  for (int i = 0; i < NRG; ++i) {
    const float* Wi = W_ff + (size_t)i * NN * NN;
    const float* Wn = W_ff + (size_t)((i + 1 < NRG) ? i + 1 : i) * NN * NN;
    float*       Wo = W_out + (size_t)i * NN * NN;
    k_pre      <<<64,  256, 0, stream>>>(A, x_inp, rbar_pre, vs_ws, x_pad, rb, i);
    k_gemm_pred<<<8,   256, 0, stream>>>(x_pad, Wi, Wn, pred);
    k_elem     <<<32,  256, 0, stream>>>(pred, v_b, v_a, ebar, vs_ws, eb, sp_out, Fpart, i);
    k_wupd     <<<512, 256, 0, stream>>>(rb, eb, Wi, Wo);
  }
  k_fin<<<1, 256, 0, stream>>>(Fpart, F_out);
}